// Sparse_svm2_42795054137419
// MI455X (gfx1250) — compile-verified
//
#include <hip/hip_runtime.h>
#include <hip/hip_bf16.h>

#define NN 16384
#define DD 64
#define NT (NN / 16)                     // 1024 j-tiles
#define JSPLIT 4
#define TILES_PER_SPLIT (NT / JSPLIT)    // 256 tiles per block (even)
#define WAVES_PER_BLOCK 8
#define ROWS_PER_BLOCK (WAVES_PER_BLOCK * 16)  // 128 rows of i per block

typedef __attribute__((ext_vector_type(2))) float v2f;
typedef __attribute__((ext_vector_type(8))) float v8f;

// ---------------------------------------------------------------------------
// Zero the s accumulator (workspace) — required each call (atomics accumulate).
// ---------------------------------------------------------------------------
__global__ void svm_zero_s(float* __restrict__ s) {
    int i = blockIdx.x * blockDim.x + threadIdx.x;
    if (i < NN) s[i] = 0.0f;
}

// Load a 16x64 f32 WMMA fragment (A and B have identical per-lane layouts for
// the Gram-matrix mapping): lane l, row = base + (l&15), elements
// {ks*4 + 2*(l>>4), +1} for ks = 0..15.  16 x global_load_b64, 8B aligned.
__device__ __forceinline__ void load_frag16(v2f* __restrict__ f,
                                            const float* __restrict__ x,
                                            int row, int hi) {
    const float* p = x + row * DD + hi * 2;
#pragma unroll
    for (int ks = 0; ks < 16; ++ks)
        f[ks] = *reinterpret_cast<const v2f*>(p + ks * 4);
}

// Fused tile: G = A·Bᵀ over K=64 via 16 v_wmma_f32_16x16x4_f32 spread across
// FOUR independent accumulator chains (dep distance 4 -> matrix pipe stays
// full), then epilogue sacc[r] += (G[r] + coeff)^degree * v_j per lane.
template <bool DEG3>
__device__ __forceinline__ void process_tile(const v2f* __restrict__ a,
                                             const v2f* __restrict__ b,
                                             const float* __restrict__ alpha,
                                             const float* __restrict__ y,
                                             int jcol, float coeffF, int degree,
                                             float* __restrict__ sacc) {
    float vj = fmaxf(alpha[jcol], 0.0f) * y[jcol];

    v8f acc[4];
#pragma unroll
    for (int c = 0; c < 4; ++c)
        acc[c] = (v8f){0.f, 0.f, 0.f, 0.f, 0.f, 0.f, 0.f, 0.f};

#pragma unroll
    for (int ks = 0; ks < 16; ++ks) {
        acc[ks & 3] = __builtin_amdgcn_wmma_f32_16x16x4_f32(
            /*neg_a=*/false, a[ks], /*neg_b=*/false, b[ks],
            /*c_mod=*/(short)0, acc[ks & 3],
            /*reuse_a=*/false, /*reuse_b=*/false);
    }

    float t[8], p[8];
#pragma unroll
    for (int r = 0; r < 8; ++r) {
        float d = (acc[0][r] + acc[1][r]) + (acc[2][r] + acc[3][r]);
        t[r] = d + coeffF;
    }
    if (DEG3) {
#pragma unroll
        for (int r = 0; r < 8; ++r) p[r] = t[r] * t[r] * t[r];
    } else {
#pragma unroll
        for (int r = 0; r < 8; ++r) p[r] = t[r];
        for (int q = 1; q < degree; ++q) {      // uniform trip, cold path only
#pragma unroll
            for (int r = 0; r < 8; ++r) p[r] *= t[r];
        }
    }
#pragma unroll
    for (int r = 0; r < 8; ++r) sacc[r] = fmaf(p[r], vj, sacc[r]);
}

// Double-buffered j-loop, templated so the degree==3 fast path has no runtime
// inner loop and no branches around the WMMA chain.
template <bool DEG3>
__device__ __forceinline__ void j_loop(const v2f* __restrict__ a,
                                       const float* __restrict__ x,
                                       const float* __restrict__ alpha,
                                       const float* __restrict__ y,
                                       int jt0, int jt1, int n, int hi,
                                       float coeffF, int degree,
                                       float* __restrict__ sacc) {
    v2f bA[16], bB[16];
    load_frag16(bA, x, jt0 * 16 + n, hi);

    for (int jt = jt0; jt < jt1; jt += 2) {
        load_frag16(bB, x, (jt + 1) * 16 + n, hi);            // prefetch next
        process_tile<DEG3>(a, bA, alpha, y, jt * 16 + n, coeffF, degree, sacc);
        if (jt + 2 < jt1)                                     // scalar branch
            load_frag16(bA, x, (jt + 2) * 16 + n, hi);
        process_tile<DEG3>(a, bB, alpha, y, (jt + 1) * 16 + n, coeffF, degree,
                           sacc);
    }
}

// ---------------------------------------------------------------------------
// Main kernel: s_i += sum_j (x_i·x_j + coeff)^degree * relu(alpha_j)*y_j
// over this block's j-range.  One 16-row i-tile per wave, A resident in VGPRs.
// ---------------------------------------------------------------------------
__global__ __launch_bounds__(256) void svm_gram_fused(
    const float* __restrict__ x, const float* __restrict__ y,
    const float* __restrict__ alpha, const int* __restrict__ coeffp,
    const int* __restrict__ degp, float* __restrict__ s) {
    const int lane = threadIdx.x & 31;
    const int wave = threadIdx.x >> 5;
    const int n    = lane & 15;   // tile column (B/N index, also C/D column)
    const int hi   = lane >> 4;   // half-wave selector (K-pair / M-half)
    const int i0   = blockIdx.x * ROWS_PER_BLOCK + wave * 16;

    const float coeffF = (float)coeffp[0];
    const int   degree = degp[0];

    // A fragments for this wave's 16 rows: 32 VGPRs, loaded once.
    v2f a[16];
    load_frag16(a, x, i0 + n, hi);

    float sacc[8];
#pragma unroll
    for (int r = 0; r < 8; ++r) sacc[r] = 0.0f;

    const int jt0 = blockIdx.y * TILES_PER_SPLIT;
    const int jt1 = jt0 + TILES_PER_SPLIT;

    if (degree == 3)
        j_loop<true>(a, x, alpha, y, jt0, jt1, n, hi, coeffF, 3, sacc);
    else
        j_loop<false>(a, x, alpha, y, jt0, jt1, n, hi, coeffF, degree, sacc);

    // Row m = r + 8*hi partials live across lanes [hi*16, hi*16+15].
    // Butterfly-reduce within each 16-lane half, then lanes 0 & 16 commit.
#pragma unroll
    for (int r = 0; r < 8; ++r) {
        float v = sacc[r];
        v += __shfl_xor(v, 1, 32);
        v += __shfl_xor(v, 2, 32);
        v += __shfl_xor(v, 4, 32);
        v += __shfl_xor(v, 8, 32);
        sacc[r] = v;
    }
    if (n == 0) {
#pragma unroll
        for (int r = 0; r < 8; ++r)
            atomicAdd(&s[i0 + hi * 8 + r], sacc[r]);
    }
}

// ---------------------------------------------------------------------------
// Final scalar loss:
//   loss = 0.5*sum(relu(alpha)) + C*sum(relu(xi))
//        + lambd * mean( y*(s+b) - (1 - relu(xi)) )
// ---------------------------------------------------------------------------
__global__ __launch_bounds__(256) void svm_loss_reduce(
    const float* __restrict__ s, const float* __restrict__ y,
    const float* __restrict__ alpha, const float* __restrict__ xi,
    const float* __restrict__ bp, const int* __restrict__ Cp,
    const int* __restrict__ lambdp, float* __restrict__ out) {
    __shared__ float shA[256], shX[256], shM[256];
    const int tid = threadIdx.x;
    const float bb = bp[0];

    float asum = 0.f, xsum = 0.f, msum = 0.f;
    for (int i = tid; i < NN; i += 256) {
        float av = fmaxf(alpha[i], 0.0f);
        float xr = fmaxf(xi[i], 0.0f);
        asum += av;
        xsum += xr;
        msum += y[i] * (s[i] + bb) - (1.0f - xr);
    }
    shA[tid] = asum; shX[tid] = xsum; shM[tid] = msum;
    __syncthreads();
    for (int off = 128; off > 0; off >>= 1) {
        if (tid < off) {
            shA[tid] += shA[tid + off];
            shX[tid] += shX[tid + off];
            shM[tid] += shM[tid + off];
        }
        __syncthreads();
    }
    if (tid == 0) {
        float Cf = (float)Cp[0];
        float lf = (float)lambdp[0];
        out[0] = 0.5f * shA[0] + Cf * shX[0] + lf * (shM[0] / (float)NN);
    }
}

// ---------------------------------------------------------------------------
extern "C" void kernel_launch(void* const* d_in, const int* in_sizes, int n_in,
                              void* d_out, int out_size, void* d_ws, size_t ws_size,
                              hipStream_t stream) {
    const float* x      = (const float*)d_in[0];
    const float* y      = (const float*)d_in[1];
    const float* alpha  = (const float*)d_in[2];
    const float* xi     = (const float*)d_in[3];
    const float* b      = (const float*)d_in[4];
    const int*   coeff  = (const int*)d_in[5];
    const int*   degree = (const int*)d_in[6];
    const int*   Cc     = (const int*)d_in[7];
    const int*   lambd  = (const int*)d_in[8];
    float* out = (float*)d_out;
    float* s   = (float*)d_ws;   // N floats = 64 KB scratch for s_i

    svm_zero_s<<<NN / 256, 256, 0, stream>>>(s);

    dim3 grid(NN / ROWS_PER_BLOCK, JSPLIT);   // 128 x 4 blocks, 8 waves each
    svm_gram_fused<<<grid, 256, 0, stream>>>(x, y, alpha, coeff, degree, s);

    svm_loss_reduce<<<1, 256, 0, stream>>>(s, y, alpha, xi, b, Cc, lambd, out);
}